// GAT_53386443489918
// MI455X (gfx1250) — compile-verified
//
#include <hip/hip_runtime.h>

#define NN    10000
#define F_IN  128
#define HID   128
#define HEADS 16
#define CLS   40
#define HF    (HEADS * HID)   // 2048

typedef __attribute__((ext_vector_type(2))) float v2f;
typedef __attribute__((ext_vector_type(8))) float v8f;

// ---- monotonic float <-> int key (for segment-max via integer atomicMax) ----
__device__ __forceinline__ int fkey(float f) {
    int b = __float_as_int(f);
    return b >= 0 ? b : (b ^ 0x7fffffff);
}
__device__ __forceinline__ float funkey(int k) {
    int b = k >= 0 ? k : (k ^ 0x7fffffff);
    return __int_as_float(b);
}

// ------------------------------- init fills ---------------------------------
__global__ void fill_f32(float* __restrict__ p, int n, float v) {
    int i = blockIdx.x * blockDim.x + threadIdx.x;
    if (i < n) p[i] = v;
}
__global__ void fill_i32(int* __restrict__ p, int n, int v) {
    int i = blockIdx.x * blockDim.x + threadIdx.x;
    if (i < n) p[i] = v;
}

// ------------------------ GEMM1: h1 = x @ W1 + b1 ---------------------------
// One wave computes a 64x16 C block (4 M-subtiles) so each strided B fetch is
// reused by 4 WMMAs.  M=10000 -> 157 groups of 64 rows (ragged tail clamped),
// N=2048 (128 tiles), K=128.
__global__ __launch_bounds__(256) void gemm1_wmma(
    const float* __restrict__ A,    // [NN, F_IN]
    const float* __restrict__ B,    // [F_IN, HF]
    const float* __restrict__ bias, // [HF]
    float* __restrict__ C)          // [NN, HF]
{
    const int TN = HF / 16;                      // 128
    int wave = threadIdx.x >> 5;
    int lane = threadIdx.x & 31;
    int job  = blockIdx.x * 8 + wave;            // 157*128 = 20096 jobs, grid=2512
    int tmg = job / TN, tn = job % TN;
    int n0 = tn * 16;
    int half = lane >> 4, l16 = lane & 15;
    int bcol = n0 + l16;

    // A-row pointers for the 4 M-subtiles; clamp ragged-tail rows in-bounds
    const float2* Ar[4];
    #pragma unroll
    for (int s = 0; s < 4; ++s) {
        int r = tmg * 64 + s * 16 + l16;
        if (r > NN - 1) r = NN - 1;
        Ar[s] = (const float2*)(A + (size_t)r * F_IN);
    }

    v8f acc[4] = {};
    #pragma unroll 4
    for (int k0 = 0; k0 < F_IN; k0 += 4) {
        int kr = k0 + 2 * half;                  // this half-wave's K pair
        v2f b;
        b.x = B[(size_t)kr * HF + bcol];         // B[kr][n]   (reused 4x)
        b.y = B[(size_t)(kr + 1) * HF + bcol];   // B[kr+1][n]
        #pragma unroll
        for (int s = 0; s < 4; ++s) {
            float2 av = Ar[s][kr >> 1];
            v2f a; a.x = av.x; a.y = av.y;
            acc[s] = __builtin_amdgcn_wmma_f32_16x16x4_f32(
                false, a, false, b, (short)0, acc[s], false, false);
        }
    }
    float bv = bias[bcol];
    #pragma unroll
    for (int s = 0; s < 4; ++s) {
        #pragma unroll
        for (int r = 0; r < 8; ++r) {
            int row = tmg * 64 + s * 16 + r + 8 * half;
            if (row < NN) C[(size_t)row * HF + bcol] = acc[s][r] + bv;
        }
    }
}

// ---------------- GEMM2: h2 = relu(out1) @ W2  (N=40 pad->48) ---------------
// Same 64x16 register blocking; ReLU fused into the A-operand load.
__global__ __launch_bounds__(256) void gemm2_wmma(
    const float* __restrict__ A,   // out1 [NN, HF]
    const float* __restrict__ B,   // W2 [HF, CLS]
    float* __restrict__ C)         // h2 [NN, CLS]
{
    const int TN = 3;                            // ceil(40/16)
    const int MG = (NN / 16 + 3) / 4;            // 157
    const int JOBS = MG * TN;                    // 471
    int wave = threadIdx.x >> 5;
    int lane = threadIdx.x & 31;
    int job  = blockIdx.x * 8 + wave;
    if (job >= JOBS) return;
    int tmg = job / TN, tn = job % TN;
    int n0 = tn * 16;
    int half = lane >> 4, l16 = lane & 15;
    int bcol = n0 + l16;
    bool colok = bcol < CLS;

    const float2* Ar[4];
    #pragma unroll
    for (int s = 0; s < 4; ++s) {
        int r = tmg * 64 + s * 16 + l16;
        if (r > NN - 1) r = NN - 1;
        Ar[s] = (const float2*)(A + (size_t)r * HF);
    }

    v8f acc[4] = {};
    #pragma unroll 2
    for (int k0 = 0; k0 < HF; k0 += 4) {
        int kr = k0 + 2 * half;
        v2f b;
        b.x = colok ? B[(size_t)kr * CLS + bcol] : 0.f;
        b.y = colok ? B[(size_t)(kr + 1) * CLS + bcol] : 0.f;
        #pragma unroll
        for (int s = 0; s < 4; ++s) {
            float2 av = Ar[s][kr >> 1];
            v2f a; a.x = fmaxf(av.x, 0.f); a.y = fmaxf(av.y, 0.f);  // fused ReLU
            acc[s] = __builtin_amdgcn_wmma_f32_16x16x4_f32(
                false, a, false, b, (short)0, acc[s], false, false);
        }
    }
    if (colok) {
        #pragma unroll
        for (int s = 0; s < 4; ++s) {
            #pragma unroll
            for (int r = 0; r < 8; ++r) {
                int row = tmg * 64 + s * 16 + r + 8 * half;
                if (row < NN) C[(size_t)row * CLS + bcol] = acc[s][r];
            }
        }
    }
}

// ------------- per-(node,head) attention dots, layer 1 (len-128) ------------
__global__ __launch_bounds__(256) void attn_node1(
    const float* __restrict__ h,        // [NN, HF]
    const float* __restrict__ att_src,  // [HEADS, HID]
    const float* __restrict__ att_dst,  // [HEADS, HID]
    float* __restrict__ as_, float* __restrict__ ad_)  // [NN, HEADS]
{
    int wave = threadIdx.x >> 5, lane = threadIdx.x & 31;
    int p = blockIdx.x * 8 + wave;      // NN*HEADS pairs, grid = 20000
    int n = p >> 4, hh = p & 15;
    const float4* hv = (const float4*)(h + (size_t)n * HF + hh * HID);
    const float4* s4 = (const float4*)(att_src + hh * HID);
    const float4* d4 = (const float4*)(att_dst + hh * HID);
    float4 x = hv[lane], s = s4[lane], d = d4[lane];
    float ss = x.x * s.x + x.y * s.y + x.z * s.z + x.w * s.w;
    float dd = x.x * d.x + x.y * d.y + x.z * d.z + x.w * d.w;
    #pragma unroll
    for (int off = 16; off; off >>= 1) {
        ss += __shfl_down(ss, off);
        dd += __shfl_down(dd, off);
    }
    if (lane == 0) { as_[p] = ss; ad_[p] = dd; }
}

// ------------- per-node attention dots, layer 2 (1 head, len-40) ------------
__global__ __launch_bounds__(256) void attn_node2(
    const float* __restrict__ h,        // [NN, CLS]
    const float* __restrict__ att_src,  // [CLS]
    const float* __restrict__ att_dst,  // [CLS]
    float* __restrict__ as_, float* __restrict__ ad_)  // [NN]
{
    int wave = threadIdx.x >> 5, lane = threadIdx.x & 31;
    int n = blockIdx.x * 8 + wave;
    if (n >= NN) return;
    const float* row = h + (size_t)n * CLS;
    float v1 = row[lane];
    float ss = v1 * att_src[lane];
    float dd = v1 * att_dst[lane];
    if (lane < CLS - 32) {
        float v2 = row[lane + 32];
        ss += v2 * att_src[lane + 32];
        dd += v2 * att_dst[lane + 32];
    }
    #pragma unroll
    for (int off = 16; off; off >>= 1) {
        ss += __shfl_down(ss, off);
        dd += __shfl_down(dd, off);
    }
    if (lane == 0) { as_[n] = ss; ad_[n] = dd; }
}

// --------- edge logits + leaky-relu + segment-max (int-keyed atomic) --------
template <int H>
__global__ void edge_logits(const int* __restrict__ ei, int E,
                            const float* __restrict__ as_, const float* __restrict__ ad_,
                            float* __restrict__ elog, int* __restrict__ mkey)
{
    int t = blockIdx.x * blockDim.x + threadIdx.x;
    int total = (E + NN) * H;
    if (t >= total) return;
    int e = t / H, hh = t % H;
    int s, d;
    if (e < E) { s = ei[e]; d = ei[E + e]; } else { s = d = e - E; }
    float v = as_[s * H + hh] + ad_[d * H + hh];
    v = v > 0.f ? v : 0.2f * v;                 // leaky relu, slope 0.2
    elog[t] = v;
    atomicMax(&mkey[d * H + hh], fkey(v));
}

// -------------------- exp(e - max) + segment-sum (atomic) -------------------
template <int H>
__global__ void edge_exp(const int* __restrict__ ei, int E,
                         const int* __restrict__ mkey,
                         float* __restrict__ elog, float* __restrict__ ssum)
{
    int t = blockIdx.x * blockDim.x + threadIdx.x;
    int total = (E + NN) * H;
    if (t >= total) return;
    int e = t / H, hh = t % H;
    int d = (e < E) ? ei[E + e] : (e - E);
    float m = funkey(mkey[d * H + hh]);
    float ex = __expf(elog[t] - m);
    elog[t] = ex;
    atomicAdd(&ssum[d * H + hh], ex);
}

// ------- layer-1 weighted scatter: out1[dst] += alpha[h] * h1[src] ---------
__global__ __launch_bounds__(256) void scatter1(
    const float* __restrict__ h1, const float* __restrict__ elog,
    const float* __restrict__ ssum, const int* __restrict__ ei, int E,
    float* __restrict__ out1)
{
    int e = blockIdx.x;
    int s, d;
    if (e < E) { s = ei[e]; d = ei[E + e]; } else { s = d = e - E; }
    __shared__ float sal[HEADS];
    if (threadIdx.x < HEADS)
        sal[threadIdx.x] = elog[e * HEADS + threadIdx.x] /
                           (ssum[d * HEADS + threadIdx.x] + 1e-16f);
    __syncthreads();
    const float* hr = h1 + (size_t)s * HF;
    float* orow = out1 + (size_t)d * HF;
    #pragma unroll
    for (int c = threadIdx.x; c < HF; c += 256)
        atomicAdd(&orow[c], hr[c] * sal[c >> 7]);   // c>>7 = head index
}

// ------- layer-2 weighted scatter: out[dst] += alpha * h2[src] -------------
__global__ __launch_bounds__(64) void scatter2(
    const float* __restrict__ h2, const float* __restrict__ elog,
    const float* __restrict__ ssum, const int* __restrict__ ei, int E,
    float* __restrict__ out)
{
    int e = blockIdx.x;
    int s, d;
    if (e < E) { s = ei[e]; d = ei[E + e]; } else { s = d = e - E; }
    float alpha = elog[e] / (ssum[d] + 1e-16f);
    int c = threadIdx.x;
    if (c < CLS)
        atomicAdd(&out[(size_t)d * CLS + c], alpha * h2[(size_t)s * CLS + c]);
}

__global__ void add_bias(float* __restrict__ out, const float* __restrict__ b) {
    int i = blockIdx.x * blockDim.x + threadIdx.x;
    if (i < NN * CLS) out[i] += b[i % CLS];
}

// ----------------------------------------------------------------------------
extern "C" void kernel_launch(void* const* d_in, const int* in_sizes, int n_in,
                              void* d_out, int out_size, void* d_ws, size_t ws_size,
                              hipStream_t stream) {
    const float* x   = (const float*)d_in[0];
    const int*   ei  = (const int*)d_in[1];
    const float* W1  = (const float*)d_in[2];
    const float* at1s= (const float*)d_in[3];
    const float* at1d= (const float*)d_in[4];
    const float* b1  = (const float*)d_in[5];
    const float* W2  = (const float*)d_in[6];
    const float* at2s= (const float*)d_in[7];
    const float* at2d= (const float*)d_in[8];
    const float* b2  = (const float*)d_in[9];
    float* out = (float*)d_out;

    const int E    = in_sizes[1] / 2;    // 160000
    const int Etot = E + NN;             // +self loops

    // workspace carve-out
    char* ws = (char*)d_ws;
    size_t off = 0;
    auto carve = [&](size_t bytes) -> void* {
        void* p = ws + off;
        off += (bytes + 255) & ~(size_t)255;
        return p;
    };
    float* h1   = (float*)carve(sizeof(float) * (size_t)NN * HF);
    float* o1   = (float*)carve(sizeof(float) * (size_t)NN * HF);
    float* nas1 = (float*)carve(sizeof(float) * NN * HEADS);
    float* nad1 = (float*)carve(sizeof(float) * NN * HEADS);
    int*   mk1  = (int*)  carve(sizeof(int)   * NN * HEADS);
    float* s1   = (float*)carve(sizeof(float) * NN * HEADS);
    float* el1  = (float*)carve(sizeof(float) * (size_t)Etot * HEADS);
    float* h2   = (float*)carve(sizeof(float) * NN * CLS);
    float* nas2 = (float*)carve(sizeof(float) * NN);
    float* nad2 = (float*)carve(sizeof(float) * NN);
    int*   mk2  = (int*)  carve(sizeof(int)   * NN);
    float* s2   = (float*)carve(sizeof(float) * NN);
    float* el2  = (float*)carve(sizeof(float) * Etot);

    const int INT_MIN_V = (int)0x80000000u;

    // ---- re-init accumulators every call (graph replays must be pure) ----
    fill_f32<<<((size_t)NN * HF + 255) / 256, 256, 0, stream>>>(o1, NN * HF, 0.f);
    fill_f32<<<(NN * HEADS + 255) / 256, 256, 0, stream>>>(s1, NN * HEADS, 0.f);
    fill_i32<<<(NN * HEADS + 255) / 256, 256, 0, stream>>>(mk1, NN * HEADS, INT_MIN_V);
    fill_f32<<<(NN + 255) / 256, 256, 0, stream>>>(s2, NN, 0.f);
    fill_i32<<<(NN + 255) / 256, 256, 0, stream>>>(mk2, NN, INT_MIN_V);
    fill_f32<<<(NN * CLS + 255) / 256, 256, 0, stream>>>(out, NN * CLS, 0.f);

    // ---- layer 1 ----
    {
        const int MG = (NN / 16 + 3) / 4;        // 157
        const int JOBS = MG * (HF / 16);         // 20096 (divisible by 8)
        gemm1_wmma<<<JOBS / 8, 256, 0, stream>>>(x, W1, b1, h1);
    }
    attn_node1<<<NN * HEADS / 8, 256, 0, stream>>>(h1, at1s, at1d, nas1, nad1);
    edge_logits<HEADS><<<((size_t)Etot * HEADS + 255) / 256, 256, 0, stream>>>(
        ei, E, nas1, nad1, el1, mk1);
    edge_exp<HEADS><<<((size_t)Etot * HEADS + 255) / 256, 256, 0, stream>>>(
        ei, E, mk1, el1, s1);
    scatter1<<<Etot, 256, 0, stream>>>(h1, el1, s1, ei, E, o1);

    // ---- layer 2 (relu fused into gemm2 A-load) ----
    {
        const int MG = (NN / 16 + 3) / 4;        // 157
        const int JOBS = MG * 3;                 // 471
        gemm2_wmma<<<(JOBS + 7) / 8, 256, 0, stream>>>(o1, W2, h2);
    }
    attn_node2<<<(NN + 7) / 8, 256, 0, stream>>>(h2, at2s, at2d, nas2, nad2);
    edge_logits<1><<<(Etot + 255) / 256, 256, 0, stream>>>(ei, E, nas2, nad2, el2, mk2);
    edge_exp<1><<<(Etot + 255) / 256, 256, 0, stream>>>(ei, E, mk2, el2, s2);
    scatter2<<<Etot, 64, 0, stream>>>(h2, el2, s2, ei, E, out);
    add_bias<<<(NN * CLS + 255) / 256, 256, 0, stream>>>(out, b2);
}